// Gate_5265629905210
// MI455X (gfx1250) — compile-verified
//
#include <hip/hip_runtime.h>

typedef __attribute__((ext_vector_type(2))) float v2f;
typedef __attribute__((ext_vector_type(8))) float v8f;
typedef unsigned long long u64;
typedef unsigned int u32;

#define BROWS   65536
#define KDIM    512
#define NEXP    64
#define NGRP    4           // 64 experts / 16 per WMMA tile
#define WAVES   8           // waves per block
#define ROWS_PER_WAVE  16
#define ROWS_PER_BLOCK (WAVES * ROWS_PER_WAVE)   // 128

// Monotone float -> u32 key (no NaNs in router scores): order-preserving.
__device__ __forceinline__ u32 f2ord(float f) {
    u32 b = __float_as_uint(f);
    u32 m = (u32)((int)b >> 31) | 0x80000000u;
    return b ^ m;
}
__device__ __forceinline__ float ord2f(u32 k) {
    u32 b = (k & 0x80000000u) ? (k ^ 0x80000000u) : ~k;
    return __uint_as_float(b);
}
// Pack (score, expert): high 32 = ordered float, low bits = 63-expert so that
// equal scores prefer the SMALLER expert index (matches jax.lax.top_k).
__device__ __forceinline__ u64 mkkey(float f, int e) {
    return ((u64)f2ord(f) << 32) | (u32)(63 - e);
}

__launch_bounds__(256)
__global__ void gate_router_kernel(const float* __restrict__ x,
                                   const float* __restrict__ w,
                                   float* __restrict__ out_w,
                                   int*   __restrict__ out_i)
{
    // Weight pre-packed as B-fragments: [kstep(128)][group(4)][lane(32)][2 floats]
    __shared__ float lds_w[NEXP * KDIM];   // 131072 bytes

    const int tid  = threadIdx.x;
    const int lane = tid & 31;
    const int wave = tid >> 5;
    const int n    = lane & 15;   // column within 16-wide tile
    const int h    = lane >> 4;   // half-wave select

    // ---- cooperative load: global W -> fragment-packed LDS ----
    for (int idx = tid; idx < NEXP * (KDIM / 4); idx += 256) {
        const int e  = idx & 63;
        const int kv = idx >> 6;
        const float4 wv = *(const float4*)(w + (size_t)e * KDIM + kv * 4);
        const int g  = e >> 4;
        const int nn = e & 15;
        float* base = lds_w + (size_t)(kv * NGRP + g) * 64;  // 32 lanes * 2 floats
        *(float2*)(base + nn * 2)        = make_float2(wv.x, wv.y);  // K = 4kv, 4kv+1
        *(float2*)(base + (16 + nn) * 2) = make_float2(wv.z, wv.w);  // K = 4kv+2, 4kv+3
    }
    __syncthreads();

    // ---- WMMA main loop: 16 rows x 64 experts per wave ----
    const int row0 = blockIdx.x * ROWS_PER_BLOCK + wave * ROWS_PER_WAVE;
    // A fragment: lane(h,n) holds x[row0+n][k0 + 2h + {0,1}]  (ISA 16x4 layout)
    const float* xa = x + (size_t)(row0 + n) * KDIM + h * 2;
    const float* lb = lds_w + lane * 2;

    v8f acc0 = {}, acc1 = {}, acc2 = {}, acc3 = {};

#pragma unroll 4
    for (int ks = 0; ks < KDIM / 4; ++ks) {
        const float2 av = *(const float2*)(xa + ks * 4);
        v2f a; a.x = av.x; a.y = av.y;

        const float* bb = lb + (size_t)ks * (NGRP * 64);
        v2f b0 = *(const v2f*)(bb + 0 * 64);
        v2f b1 = *(const v2f*)(bb + 1 * 64);
        v2f b2 = *(const v2f*)(bb + 2 * 64);
        v2f b3 = *(const v2f*)(bb + 3 * 64);

        acc0 = __builtin_amdgcn_wmma_f32_16x16x4_f32(false, a, false, b0, (short)0, acc0, false, false);
        acc1 = __builtin_amdgcn_wmma_f32_16x16x4_f32(false, a, false, b1, (short)0, acc1, false, false);
        acc2 = __builtin_amdgcn_wmma_f32_16x16x4_f32(false, a, false, b2, (short)0, acc2, false, false);
        acc3 = __builtin_amdgcn_wmma_f32_16x16x4_f32(false, a, false, b3, (short)0, acc3, false, false);
    }

    // ---- fused softmax + top-2 epilogue (branch-free, select-based) ----
    // C layout: VGPR r -> row r (lanes 0-15) / row r+8 (lanes 16-31); N = n.
#pragma unroll
    for (int r = 0; r < 8; ++r) {
        const float s0 = acc0[r], s1 = acc1[r], s2 = acc2[r], s3 = acc3[r];

        u64 v1 = mkkey(s0, n);
        u64 v2 = 0;

        // branch-free top-2 insert: pure u64 compare + select (v_cndmask)
        #define INS2(c) do { u64 _c = (c); bool _g = _c > v1;       \
                             u64 _hi = _g ? _c : v1;                \
                             u64 _lo = _g ? v1 : _c;                \
                             v1 = _hi;                              \
                             v2 = (_lo > v2) ? _lo : v2; } while (0)

        INS2(mkkey(s1, 16 + n));
        INS2(mkkey(s2, 32 + n));
        INS2(mkkey(s3, 48 + n));

        // merge top-2 pairs across the 16 lanes of this half-wave
        // (xor masks 1/2/4/8 never cross the lane-16 boundary on wave32)
#pragma unroll
        for (int m = 1; m <= 8; m <<= 1) {
            u64 ov1 = __shfl_xor(v1, m, 32);
            u64 ov2 = __shfl_xor(v2, m, 32);
            INS2(ov1);
            INS2(ov2);
        }
        #undef INS2

        const float m1 = ord2f((u32)(v1 >> 32));
        const float m2 = ord2f((u32)(v2 >> 32));
        const int   e1 = 63 - (int)(v1 & 0x3Fu);
        const int   e2 = 63 - (int)(v2 & 0x3Fu);

        // softmax denominator relative to row max m1
        float z = __expf(s0 - m1) + __expf(s1 - m1) + __expf(s2 - m1) + __expf(s3 - m1);
#pragma unroll
        for (int m = 1; m <= 8; m <<= 1)
            z += __shfl_xor(z, m, 32);

        if (n == 0) {
            const int row = row0 + r + 8 * h;
            const float inv = 1.0f / z;                  // exp(m1-m1)/Z
            const float w2v = __expf(m2 - m1) * inv;     // exp(m2-m1)/Z
            *(float2*)(out_w + (size_t)row * 2) = make_float2(inv, w2v);
            *(int2*)(out_i + (size_t)row * 2)   = make_int2(e1, e2);
        }
    }
}

extern "C" void kernel_launch(void* const* d_in, const int* in_sizes, int n_in,
                              void* d_out, int out_size, void* d_ws, size_t ws_size,
                              hipStream_t stream) {
    const float* x = (const float*)d_in[0];     // [65536, 512] fp32
    const float* w = (const float*)d_in[1];     // [64, 512] fp32
    float* out_w = (float*)d_out;                               // [65536, 2] fp32
    int*   out_i = (int*)((float*)d_out + (size_t)BROWS * 2);   // [65536, 2] i32

    dim3 grid(BROWS / ROWS_PER_BLOCK);  // 512 blocks
    dim3 block(256);                    // 8 waves (wave32)
    gate_router_kernel<<<grid, block, 0, stream>>>(x, w, out_w, out_i);
}